// PerPositionEMADelta_35588099014907
// MI455X (gfx1250) — compile-verified
//
#include <hip/hip_runtime.h>
#include <hip/hip_bf16.h>
#include <math.h>

// Problem constants (match reference)
#define Bq 32768
#define Lq 24
#define Hq 64
#define Vq 64

#define WAVES_PER_BLOCK 2
#define BLOCK_THREADS (WAVES_PER_BLOCK * 32)

typedef __attribute__((ext_vector_type(2))) float v2f;
typedef __attribute__((ext_vector_type(4))) float v4f;
typedef __attribute__((ext_vector_type(8))) float v8f;

// ---- LDS layout (float offsets) ----
// Weight matrices stored QUAD-INTERLEAVED: element (k,n) of a KxN matrix at
//   [(k>>2)*(4N) + 4n + (k&3)]
// so each WMMA B-fragment (two consecutive k, fixed n) is one contiguous v2f
// -> single conflict-free ds_load_b64 per operand.
#define OFF_W1   0            // [64][128] quad-packed (8192 floats)
#define OFF_W2   8192         // [128][64] quad-packed (8192)
#define OFF_RP   16384        // [64][64]  quad-packed (4096)
#define OFF_OW   20480        // [64][64]  quad-packed (4096)
#define OFF_B1   24576        // [128]
#define OFF_B2   24704        // [64]
#define OFF_LNG  24768        // [64]
#define OFF_LNB  24832        // [64]
#define OFF_RPB  24896        // [64]
#define OFF_OB   24960        // [64]
#define OFF_OM   25024        // [32] holds (1 - alpha_t)
#define OFF_WAVE 25056
// per-wave private region
#define H_STRIDE 68           // 64 + 4 pad -> conflict-free b64 A reads
#define HBUF_FLOATS (Lq * 16 * H_STRIDE)      // 26112
#define X1_STRIDE 132
#define X1_FLOATS (16 * X1_STRIDE)            // 2112
#define WAVE_FLOATS (HBUF_FLOATS + X1_FLOATS) // 28224
#define SMEM_FLOATS (OFF_WAVE + WAVES_PER_BLOCK * WAVE_FLOATS) // 81504
#define SMEM_BYTES (SMEM_FLOATS * 4)          // 326016 <= 327680 (320KB/WG)

// One 16x16 output tile of A[16xK] @ B[KxN], fp32 WMMA (16x16x4).
// A fragment: lane<16 -> (k,k+1) of row lane; lane>=16 -> (k+2,k+3). One b64.
// B fragment from quad-packed weights: one b64 at (k>>2)*nstride4 + 4*col + hk.
template <int K>
__device__ inline v8f gemm_tile_f32(const float* __restrict__ As, int lda,
                                    const float* __restrict__ Bqp, int nstride4,
                                    v8f acc, int lane) {
  const int rc = lane & 15;
  const int hk = (lane >> 4) << 1;
  const float* ap = As + rc * lda + hk;
  const float* bp = Bqp + rc * 4 + hk;
#pragma unroll
  for (int k = 0; k < K; k += 4) {
    v2f a = *(const v2f*)(ap + k);
    v2f b = *(const v2f*)(bp + (k >> 2) * nstride4);
    acc = __builtin_amdgcn_wmma_f32_16x16x4_f32(
        /*neg_a=*/false, a, /*neg_b=*/false, b,
        /*c_mod=*/(short)0, acc, /*reuse_a=*/false, /*reuse_b=*/false);
  }
  return acc;
}

__global__ __launch_bounds__(BLOCK_THREADS)
void PerPositionEMADelta_fused(const int* __restrict__ seq,
                               const float* __restrict__ embed,
                               const float* __restrict__ w1,
                               const float* __restrict__ b1,
                               const float* __restrict__ w2,
                               const float* __restrict__ b2,
                               const float* __restrict__ ln_g,
                               const float* __restrict__ ln_b,
                               const float* __restrict__ rp_w,
                               const float* __restrict__ rp_b,
                               const float* __restrict__ out_w,
                               const float* __restrict__ out_b,
                               const float* __restrict__ alpha_logits,
                               float* __restrict__ out) {
  extern __shared__ float sm[];
  const int tid  = threadIdx.x;
  const int lane = tid & 31;
  const int wave = tid >> 5;
  const int rc   = lane & 15;   // row (A) or col (B/C) within a 16-tile
  const int half = lane >> 4;   // 0: lanes 0-15, 1: lanes 16-31

  float* s_w1  = sm + OFF_W1;
  float* s_w2  = sm + OFF_W2;
  float* s_rp  = sm + OFF_RP;
  float* s_ow  = sm + OFF_OW;
  float* s_b1  = sm + OFF_B1;
  float* s_b2  = sm + OFF_B2;
  float* s_lng = sm + OFF_LNG;
  float* s_lnb = sm + OFF_LNB;
  float* s_rpb = sm + OFF_RPB;
  float* s_ob  = sm + OFF_OB;
  float* s_om  = sm + OFF_OM;
  float* wbase = sm + OFF_WAVE + wave * WAVE_FLOATS;
  float* s_H   = wbase;                 // [24][16][H_STRIDE]
  float* s_X1  = wbase + HBUF_FLOATS;   // [16][X1_STRIDE]

  // ---- cooperative weight preload into LDS (quad-interleave on the fly) ----
  for (int i = tid; i < 64 * 128; i += BLOCK_THREADS) {       // w1: K=64,N=128
    int k = i >> 7, n = i & 127;
    s_w1[(k >> 2) * 512 + n * 4 + (k & 3)] = w1[i];
  }
  for (int i = tid; i < 128 * 64; i += BLOCK_THREADS) {       // w2: K=128,N=64
    int k = i >> 6, n = i & 63;
    s_w2[(k >> 2) * 256 + n * 4 + (k & 3)] = w2[i];
  }
  for (int i = tid; i < 64 * 64; i += BLOCK_THREADS) {        // rp_w / out_w
    int k = i >> 6, n = i & 63;
    int p = (k >> 2) * 256 + n * 4 + (k & 3);
    s_rp[p] = rp_w[i];
    s_ow[p] = out_w[i];
  }
  for (int i = tid; i < 128; i += BLOCK_THREADS) s_b1[i] = b1[i];
  if (tid < 64) {
    s_b2[tid]  = b2[tid];
    s_lng[tid] = ln_g[tid];
    s_lnb[tid] = ln_b[tid];
    s_rpb[tid] = rp_b[tid];
    s_ob[tid]  = out_b[tid];
  }
  if (tid < Lq) {
    float sg = 1.0f / (1.0f + expf(-alpha_logits[tid]));
    s_om[tid] = 1.0f - (0.49f * sg + 0.5f);   // (1 - alpha_t)
  }
  __syncthreads();

  const int base = (blockIdx.x * WAVES_PER_BLOCK + wave) * 16; // batch rows

  // ---- Phase 1: per-token embed gather -> MLP -> residual + LayerNorm ----
  for (int t = 0; t < Lq; ++t) {
    float* Ht = s_H + t * (16 * H_STRIDE);
    {
      int idx = seq[(size_t)(base + rc) * Lq + t];
      const float* src = embed + (size_t)idx * Hq + half * 32;
      float* dst = Ht + rc * H_STRIDE + half * 32;
#pragma unroll
      for (int j = 0; j < 32; j += 4) *(v4f*)(dst + j) = *(const v4f*)(src + j);
    }
    __syncthreads();

    // X1 = relu(H @ w1 + b1)   [16 x 128]
    for (int nt = 0; nt < 8; ++nt) {
      v8f acc;
      float bv = s_b1[nt * 16 + rc];
#pragma unroll
      for (int j = 0; j < 8; ++j) acc[j] = bv;
      acc = gemm_tile_f32<64>(Ht, H_STRIDE, s_w1 + nt * 16 * 4, 512, acc, lane);
#pragma unroll
      for (int j = 0; j < 8; ++j) {
        int row = j + 8 * half;
        s_X1[row * X1_STRIDE + nt * 16 + rc] = fmaxf(acc[j], 0.0f);
      }
    }
    __syncthreads();

    // H += X1 @ w2 + b2        [16 x 64] residual in place
    for (int nt = 0; nt < 4; ++nt) {
      v8f acc;
      float bv = s_b2[nt * 16 + rc];
#pragma unroll
      for (int j = 0; j < 8; ++j) acc[j] = bv;
      acc = gemm_tile_f32<128>(s_X1, X1_STRIDE, s_w2 + nt * 16 * 4, 256, acc, lane);
#pragma unroll
      for (int j = 0; j < 8; ++j) {
        int row = j + 8 * half;
        Ht[row * H_STRIDE + nt * 16 + rc] += acc[j];
      }
    }
    __syncthreads();

    // LayerNorm each row (lanes 0..15 own one row each; no WMMA in here)
    if (half == 0) {
      float* rowp = Ht + rc * H_STRIDE;
      float mu = 0.0f;
#pragma unroll
      for (int c = 0; c < 64; ++c) mu += rowp[c];
      mu *= (1.0f / 64.0f);
      float var = 0.0f;
#pragma unroll
      for (int c = 0; c < 64; ++c) { float d = rowp[c] - mu; var += d * d; }
      var *= (1.0f / 64.0f);
      float inv = rsqrtf(var + 1e-5f);
#pragma unroll
      for (int c = 0; c < 64; ++c)
        rowp[c] = (rowp[c] - mu) * inv * s_lng[c] + s_lnb[c];
    }
    __syncthreads();
  }

  // ---- Phase 2: backward-unrolled delta-rule scan ----
  //   M_final·h_last == sum_s (1-a_s)(kn_s·q_s) k_s  with  q updated backward.
  // Each lane holds 32 columns of row rc (half selects column block).
  float q[32], v[32];
  {
    const float* hl = s_H + 23 * (16 * H_STRIDE) + rc * H_STRIDE + half * 32;
#pragma unroll
    for (int j = 0; j < 32; ++j) { q[j] = hl[j]; v[j] = 0.0f; }
  }
  for (int s = Lq - 2; s >= 0; --s) {
    const float* hs = s_H + s * (16 * H_STRIDE) + rc * H_STRIDE + half * 32;
    float hk[32];
    float nsq = 0.0f;
#pragma unroll
    for (int j = 0; j < 32; ++j) { hk[j] = hs[j]; nsq += hk[j] * hk[j]; }
    nsq += __shfl_xor(nsq, 16, 32);                 // combine column halves
    float inv = 1.0f / fmaxf(sqrtf(nsq), 1e-12f);   // 1/max(||k||,eps)
    float dq = 0.0f;
#pragma unroll
    for (int j = 0; j < 32; ++j) dq += hk[j] * q[j];
    dq += __shfl_xor(dq, 16, 32);
    float c = s_om[s] * (dq * inv);                 // (1-a)*(kn . q)
    float ci = c * inv;
#pragma unroll
    for (int j = 0; j < 32; ++j) {
      v[j] += c * hk[j];    // + c * k
      q[j] -= ci * hk[j];   // - c * kn
    }
  }

  // ---- Phase 3: output head  out = (v @ rp_w + rp_b) @ out_w + out_b ----
  float* s_V  = s_X1;   // 16 x H_STRIDE fits inside X1 region
  float* s_T1 = s_H;    // history no longer needed
  {
    float* dst = s_V + rc * H_STRIDE + half * 32;
#pragma unroll
    for (int j = 0; j < 32; ++j) dst[j] = v[j];
  }
  __syncthreads();

  for (int nt = 0; nt < 4; ++nt) {
    v8f acc;
    float bv = s_rpb[nt * 16 + rc];
#pragma unroll
    for (int j = 0; j < 8; ++j) acc[j] = bv;
    acc = gemm_tile_f32<64>(s_V, H_STRIDE, s_rp + nt * 16 * 4, 256, acc, lane);
#pragma unroll
    for (int j = 0; j < 8; ++j) {
      int row = j + 8 * half;
      s_T1[row * H_STRIDE + nt * 16 + rc] = acc[j];
    }
  }
  __syncthreads();

  for (int nt = 0; nt < 4; ++nt) {
    v8f acc;
    float bv = s_ob[nt * 16 + rc];
#pragma unroll
    for (int j = 0; j < 8; ++j) acc[j] = bv;
    acc = gemm_tile_f32<64>(s_T1, H_STRIDE, s_ow + nt * 16 * 4, 256, acc, lane);
#pragma unroll
    for (int j = 0; j < 8; ++j) {
      int row = j + 8 * half;
      out[(size_t)(base + row) * Vq + nt * 16 + rc] = acc[j];
    }
  }
}

extern "C" void kernel_launch(void* const* d_in, const int* in_sizes, int n_in,
                              void* d_out, int out_size, void* d_ws, size_t ws_size,
                              hipStream_t stream) {
  (void)n_in; (void)out_size; (void)d_ws; (void)ws_size;
  const int* seq            = (const int*)d_in[0];
  const float* embed        = (const float*)d_in[1];
  const float* w1           = (const float*)d_in[2];
  const float* b1           = (const float*)d_in[3];
  const float* w2           = (const float*)d_in[4];
  const float* b2           = (const float*)d_in[5];
  const float* ln_g         = (const float*)d_in[6];
  const float* ln_b         = (const float*)d_in[7];
  const float* rp_w         = (const float*)d_in[8];
  const float* rp_b         = (const float*)d_in[9];
  const float* out_w        = (const float*)d_in[10];
  const float* out_b        = (const float*)d_in[11];
  const float* alpha_logits = (const float*)d_in[12];
  float* out = (float*)d_out;

  const int B = in_sizes[0] / Lq;                 // 32768
  const int blocks = B / (16 * WAVES_PER_BLOCK);  // 1024

  // Allow >48KB dynamic LDS (CDNA5 WGP: up to 320KB per workgroup)
  (void)hipFuncSetAttribute((const void*)PerPositionEMADelta_fused,
                            hipFuncAttributeMaxDynamicSharedMemorySize,
                            (int)SMEM_BYTES);

  PerPositionEMADelta_fused<<<blocks, BLOCK_THREADS, SMEM_BYTES, stream>>>(
      seq, embed, w1, b1, w2, b2, ln_g, ln_b, rp_w, rp_b, out_w, out_b,
      alpha_logits, out);
}